// GATEncoder_53163105190282
// MI455X (gfx1250) — compile-verified
//
#include <hip/hip_runtime.h>
#include <math.h>

typedef float v2f __attribute__((ext_vector_type(2)));
typedef float v8f __attribute__((ext_vector_type(8)));

__device__ __forceinline__ float leaky(float x, float s) {
  return x > 0.0f ? x : s * x;
}
// Monotonic float->uint mapping for atomicMax over signed floats.
__device__ __forceinline__ unsigned f2key(float f) {
  unsigned u = __float_as_uint(f);
  return (u & 0x80000000u) ? ~u : (u | 0x80000000u);
}
__device__ __forceinline__ float key2f(unsigned k) {
  unsigned u = (k & 0x80000000u) ? (k & 0x7fffffffu) : ~k;
  return __uint_as_float(u);
}

// ---------------- Layer-0 GEMM: [N,2] x [2,64] -> [N,64] (K=2, pure VALU) ----
__global__ __launch_bounds__(256) void gat_gemm0(const float* __restrict__ x,
                                                 const float* __restrict__ W,
                                                 float* __restrict__ h, int N) {
  int i = blockIdx.x * blockDim.x + threadIdx.x;
  if (i >= N * 16) return;
  int n = i >> 4, q = (i & 15);
  float x0 = x[(size_t)n * 2 + 0];
  float x1 = x[(size_t)n * 2 + 1];
  float4 a = ((const float4*)W)[q];        // W[0][4q..]
  float4 b = ((const float4*)(W + 64))[q]; // W[1][4q..]
  float4 o;
  o.x = x0 * a.x + x1 * b.x;
  o.y = x0 * a.y + x1 * b.y;
  o.z = x0 * a.z + x1 * b.z;
  o.w = x0 * a.w + x1 * b.w;
  *(float4*)(h + (size_t)n * 64 + q * 4) = o;
}

// ---------------- [N,64] x [64,64] GEMM via V_WMMA_F32_16X16X4_F32 ----------
// One wave computes a 16-row x 64-col strip: 4 col-tiles x 16 K-blocks of WMMA.
__global__ __launch_bounds__(256) void gat_gemm64_wmma(const float* __restrict__ X,
                                                       const float* __restrict__ W,
                                                       float* __restrict__ Hout, int N) {
  int wave = (blockIdx.x * blockDim.x + threadIdx.x) >> 5;
  int lane = threadIdx.x & 31;
  int row0 = wave * 16;
  if (row0 >= N) return;                 // wave-uniform exit; EXEC stays full
  int hl = lane >> 4;                    // half: 0 -> K {0,1}, 1 -> K {2,3}
  int l  = lane & 15;
  int rowA = row0 + l; if (rowA >= N) rowA = N - 1;  // clamp (keeps EXEC full)
  const v2f* xr2 = (const v2f*)(X + (size_t)rowA * 64);
  v2f a[16];
#pragma unroll
  for (int kb = 0; kb < 16; ++kb) a[kb] = xr2[kb * 2 + hl];
  bool fullTile = (row0 + 16 <= N);      // wave-uniform: scalar branch, no EXEC ops
#pragma unroll
  for (int nt = 0; nt < 4; ++nt) {
    int col = nt * 16 + l;
    v8f c = {0.f, 0.f, 0.f, 0.f, 0.f, 0.f, 0.f, 0.f};
#pragma unroll
    for (int kb = 0; kb < 16; ++kb) {
      int k0 = kb * 4 + hl * 2;
      v2f b;
      b[0] = W[(size_t)k0 * 64 + col];
      b[1] = W[(size_t)(k0 + 1) * 64 + col];
      c = __builtin_amdgcn_wmma_f32_16x16x4_f32(false, a[kb], false, b,
                                                (short)0, c, false, false);
    }
    // C/D layout: VGPR r holds row M = r + 8*half; stores coalesce per half-wave.
    if (fullTile) {
#pragma unroll
      for (int r = 0; r < 8; ++r)
        Hout[(size_t)(row0 + r + hl * 8) * 64 + col] = c[r];
    } else {
#pragma unroll
      for (int r = 0; r < 8; ++r) {
        int row = row0 + r + hl * 8;
        if (row < N) Hout[(size_t)row * 64 + col] = c[r];
      }
    }
  }
}

// ---------------- alpha_s / alpha_d: per (node, head) dot of length C --------
__global__ __launch_bounds__(256) void gat_alpha(const float* __restrict__ h,
                                                 const float* __restrict__ a_src,
                                                 const float* __restrict__ a_dst,
                                                 float* __restrict__ as_,
                                                 float* __restrict__ ad_,
                                                 int N, int H) {
  int i = blockIdx.x * blockDim.x + threadIdx.x;
  if (i >= N * H) return;
  int n = i / H, hh = i % H;
  int C = 64 / H;
  const float* hp = h + (size_t)n * 64 + hh * C;
  const float* sp = a_src + hh * C;
  const float* dp = a_dst + hh * C;
  float s1 = 0.f, s2 = 0.f;
  for (int c = 0; c < C; ++c) { float v = hp[c]; s1 += v * sp[c]; s2 += v * dp[c]; }
  as_[i] = s1;
  ad_[i] = s2;
}

// ---------------- zero accumulator + softmax state ---------------------------
__global__ __launch_bounds__(256) void gat_init(float* __restrict__ acc,
                                                unsigned* __restrict__ mkey,
                                                float* __restrict__ ssum,
                                                int N, int H) {
  int i = blockIdx.x * blockDim.x + threadIdx.x;
  if (i >= N * 64) return;
  acc[i] = 0.0f;
  if (i < N * H) { mkey[i] = 0u; ssum[i] = 0.0f; }   // key 0 == max identity
}

// ---------------- pass A: segment max of logits over dst ---------------------
__global__ __launch_bounds__(256) void gat_edge_max(const long long* __restrict__ ei,
                                                    const float* __restrict__ as_,
                                                    const float* __restrict__ ad_,
                                                    unsigned* __restrict__ mkey,
                                                    int N, int E, int H) {
  int e = blockIdx.x * blockDim.x + threadIdx.x;
  int ET = E + N;                     // self-loops appended
  if (e >= ET) return;
  int s, d;
  if (e < E) { s = (int)ei[e]; d = (int)ei[(size_t)E + e]; }
  else       { s = d = e - E; }
  for (int hh = 0; hh < H; ++hh) {
    float lg = leaky(as_[(size_t)s * H + hh] + ad_[(size_t)d * H + hh], 0.2f);
    atomicMax(&mkey[(size_t)d * H + hh], f2key(lg));
  }
}

// ---------------- pass B: fused exp + segment-sum + weighted scatter ---------
// 16 threads per edge; thread t handles 4 channels [4t, 4t+4).
__global__ __launch_bounds__(256) void gat_edge_scatter(const long long* __restrict__ ei,
                                                        const float* __restrict__ as_,
                                                        const float* __restrict__ ad_,
                                                        const unsigned* __restrict__ mkey,
                                                        float* __restrict__ ssum,
                                                        const float* __restrict__ h,
                                                        float* __restrict__ acc,
                                                        int N, int E, int H) {
  int tid = blockIdx.x * blockDim.x + threadIdx.x;
  int ET = E + N;
  if (tid >= ET * 16) return;
  int e = tid >> 4;
  int t = tid & 15;
  int s, d;
  if (e < E) { s = (int)ei[e]; d = (int)ei[(size_t)E + e]; }
  else       { s = d = e - E; }
  int C  = 64 / H;
  int c0 = t * 4;
  int hh = c0 / C;
  float lg = leaky(as_[(size_t)s * H + hh] + ad_[(size_t)d * H + hh], 0.2f);
  float m  = key2f(mkey[(size_t)d * H + hh]);
  float w  = __expf(lg - m);
  if ((c0 % C) == 0) atomicAdd(&ssum[(size_t)d * H + hh], w);
  float4 hv = *(const float4*)(h + (size_t)s * 64 + c0);
  float* ap = acc + (size_t)d * 64 + c0;
  atomicAdd(ap + 0, w * hv.x);
  atomicAdd(ap + 1, w * hv.y);
  atomicAdd(ap + 2, w * hv.z);
  atomicAdd(ap + 3, w * hv.w);
}

// ---------------- finalize: normalize, +bias, optional leaky_relu ------------
__global__ __launch_bounds__(256) void gat_finalize(const float* __restrict__ acc,
                                                    const float* __restrict__ ssum,
                                                    const float* __restrict__ bias,
                                                    float* __restrict__ outp,
                                                    int N, int H, int act) {
  int i = blockIdx.x * blockDim.x + threadIdx.x;
  if (i >= N * 16) return;
  int n  = i >> 4;
  int c0 = (i & 15) * 4;
  int C  = 64 / H;
  int hh = c0 / C;
  float inv = 1.0f / ssum[(size_t)n * H + hh];
  float4 a = *(const float4*)(acc + (size_t)n * 64 + c0);
  float4 b = *(const float4*)(bias + c0);
  float4 o;
  o.x = a.x * inv + b.x;
  o.y = a.y * inv + b.y;
  o.z = a.z * inv + b.z;
  o.w = a.w * inv + b.w;
  if (act) {
    o.x = leaky(o.x, 0.01f); o.y = leaky(o.y, 0.01f);
    o.z = leaky(o.z, 0.01f); o.w = leaky(o.w, 0.01f);
  }
  *(float4*)(outp + (size_t)n * 64 + c0) = o;
}

static inline int cdiv_ll(long long a, long long b) { return (int)((a + b - 1) / b); }

extern "C" void kernel_launch(void* const* d_in, const int* in_sizes, int n_in,
                              void* d_out, int out_size, void* d_ws, size_t ws_size,
                              hipStream_t stream) {
  const float*     x   = (const float*)d_in[0];
  const long long* ei  = (const long long*)d_in[1];
  const float* W0  = (const float*)d_in[2];
  const float* as0 = (const float*)d_in[3];
  const float* ad0 = (const float*)d_in[4];
  const float* b0  = (const float*)d_in[5];
  const float* W1  = (const float*)d_in[6];
  const float* as1 = (const float*)d_in[7];
  const float* ad1 = (const float*)d_in[8];
  const float* b1  = (const float*)d_in[9];
  const float* W2  = (const float*)d_in[10];
  const float* as2 = (const float*)d_in[11];
  const float* ad2 = (const float*)d_in[12];
  const float* b2  = (const float*)d_in[13];

  int N = in_sizes[0] / 2;   // x: [N,2]
  int E = in_sizes[1] / 2;   // edge_index: [2,E]
  int ET = E + N;

  // Workspace layout (d_out doubles as a ping-pong feature buffer).
  size_t NB = (size_t)N * 64;
  float*    bufh = (float*)d_ws;          // GEMM output h
  float*    buf2 = bufh + NB;             // layer-1 output features
  float*    asb  = buf2 + NB;             // alpha_s  [N, H]
  float*    adb  = asb + (size_t)N * 4;   // alpha_d  [N, H]
  unsigned* mkey = (unsigned*)(adb + (size_t)N * 4);  // segment-max keys
  float*    ssum = (float*)(mkey + (size_t)N * 4);    // segment sums
  float*    fout = (float*)d_out;

  const int B = 256;
  int gN16   = cdiv_ll((long long)N * 16, B);
  int gNH4   = cdiv_ll((long long)N * 4, B);
  int gN64   = cdiv_ll((long long)N * 64, B);
  int gE     = cdiv_ll(ET, B);
  int gE16   = cdiv_ll((long long)ET * 16, B);
  int gGemm  = cdiv_ll((long long)((N + 15) / 16) * 32, B);  // 1 wave / 16 rows

  // ---------------- layer 0: H=4, in = x (K=2), acc/out = d_out, leaky 0.01
  gat_gemm0<<<gN16, B, 0, stream>>>(x, W0, bufh, N);
  gat_alpha<<<gNH4, B, 0, stream>>>(bufh, as0, ad0, asb, adb, N, 4);
  gat_init<<<gN64, B, 0, stream>>>(fout, mkey, ssum, N, 4);
  gat_edge_max<<<gE, B, 0, stream>>>(ei, asb, adb, mkey, N, E, 4);
  gat_edge_scatter<<<gE16, B, 0, stream>>>(ei, asb, adb, mkey, ssum, bufh, fout, N, E, 4);
  gat_finalize<<<gN16, B, 0, stream>>>(fout, ssum, b0, fout, N, 4, 1);

  // ---------------- layer 1: H=4, in = d_out, acc/out = buf2, leaky 0.01
  gat_gemm64_wmma<<<gGemm, B, 0, stream>>>(fout, W1, bufh, N);
  gat_alpha<<<gNH4, B, 0, stream>>>(bufh, as1, ad1, asb, adb, N, 4);
  gat_init<<<gN64, B, 0, stream>>>(buf2, mkey, ssum, N, 4);
  gat_edge_max<<<gE, B, 0, stream>>>(ei, asb, adb, mkey, N, E, 4);
  gat_edge_scatter<<<gE16, B, 0, stream>>>(ei, asb, adb, mkey, ssum, bufh, buf2, N, E, 4);
  gat_finalize<<<gN16, B, 0, stream>>>(buf2, ssum, b1, buf2, N, 4, 1);

  // ---------------- layer 2: H=1, in = buf2, acc/out = d_out, no activation
  gat_gemm64_wmma<<<gGemm, B, 0, stream>>>(buf2, W2, bufh, N);
  gat_alpha<<<cdiv_ll(N, B), B, 0, stream>>>(bufh, as2, ad2, asb, adb, N, 1);
  gat_init<<<gN64, B, 0, stream>>>(fout, mkey, ssum, N, 1);
  gat_edge_max<<<gE, B, 0, stream>>>(ei, asb, adb, mkey, N, E, 1);
  gat_edge_scatter<<<gE16, B, 0, stream>>>(ei, asb, adb, mkey, ssum, bufh, fout, N, E, 1);
  gat_finalize<<<gN16, B, 0, stream>>>(fout, ssum, b2, fout, N, 1, 0);
}